// TypeLayer_53979148976539
// MI455X (gfx1250) — compile-verified
//
#include <hip/hip_runtime.h>

typedef __attribute__((ext_vector_type(2))) float v2f;
typedef __attribute__((ext_vector_type(8))) float v8f;

#define N_RELV 2000
#define DIM    128
#define CLIPF  1.0e6f

// ---------------------------------------------------------------------------
// Kernel 1: zero the output accumulator (must be re-zeroed every call).
// ---------------------------------------------------------------------------
__global__ void zero_out(float4* __restrict__ out, int n4) {
  int i = blockIdx.x * blockDim.x + threadIdx.x;
  if (i < n4) out[i] = make_float4(0.f, 0.f, 0.f, 0.f);
}

// ---------------------------------------------------------------------------
// Kernel 2: T[r, :] = clip(rel_features[r, :] @ W^T + b, +-CLIP)
// One wave computes one 16x16 output tile via V_WMMA_F32_16X16X4_F32,
// looping K = 0..127 in steps of 4.
//
// A (16x4 f32, 2 VGPRs): lanes 0-15 hold K=k+0 (v0) / K=k+1 (v1) for M=lane;
//                        lanes 16-31 hold K=k+2 / K=k+3 for M=lane-16.
// B (4x16 f32, 2 VGPRs): mirrored (N=lane%16, K offset by (lane>>4)*2).
// Since B = W^T (B[k][n] = W[n][k]), A and B share the same gather pattern.
// C/D (16x16 f32, 8 VGPRs): VGPR j holds M=j (lanes 0-15) / M=j+8 (lanes 16-31),
//                           N = lane%16.
// ---------------------------------------------------------------------------
__global__ __launch_bounds__(128) void rel_gemm(const float* __restrict__ rel,
                                                const float* __restrict__ W,
                                                const float* __restrict__ bias,
                                                float* __restrict__ T) {
  const int wave   = (blockIdx.x * blockDim.x + threadIdx.x) >> 5;
  const int lane   = threadIdx.x & 31;
  const int tile_m = wave >> 3;   // 0..124  (2000/16 row tiles)
  const int tile_n = wave & 7;    // 0..7    (128/16 col tiles)
  if (tile_m >= N_RELV / 16) return;  // wave-uniform; never taken at this grid

  const int row0 = tile_m * 16;
  const int col0 = tile_n * 16;
  const int mn   = lane & 15;
  const int koff = (lane >> 4) << 1;

  const float* __restrict__ arow = rel + (size_t)(row0 + mn) * DIM + koff;
  const float* __restrict__ brow = W   + (size_t)(col0 + mn) * DIM + koff;

  v8f c = {};
#pragma unroll
  for (int k = 0; k < DIM; k += 4) {
    v2f a;  a.x  = arow[k]; a.y  = arow[k + 1];
    v2f bb; bb.x = brow[k]; bb.y = brow[k + 1];
    // 8 args: (neg_a, A, neg_b, B, c_mod, C, reuse_a, reuse_b)
    c = __builtin_amdgcn_wmma_f32_16x16x4_f32(false, a, false, bb,
                                              (short)0, c, false, false);
  }

  const int   n     = col0 + mn;
  const float bv    = bias[n];
  const int   mbase = (lane >> 4) << 3;
#pragma unroll
  for (int j = 0; j < 8; ++j) {
    float v = c[j] + bv;
    v = fminf(fmaxf(v, -CLIPF), CLIPF);
    T[(size_t)(row0 + mbase + j) * DIM + n] = v;
  }
}

// ---------------------------------------------------------------------------
// Kernel 3: scatter-add. One wave per fact: 32 lanes x float4 = the full
// 128-float row of T[rel]; 4 fp32 atomics per lane per target (tail, head).
// Output accumulator (8.2 MB) is L2-resident, so these are L2 atomics.
// ---------------------------------------------------------------------------
__global__ __launch_bounds__(256) void scatter_facts(const float* __restrict__ T,
                                                     const int* __restrict__ rels,
                                                     const int* __restrict__ heads,
                                                     const int* __restrict__ tails,
                                                     float* __restrict__ out,
                                                     int num_fact) {
  const int lane   = threadIdx.x & 31;
  const int wave   = (blockIdx.x * blockDim.x + threadIdx.x) >> 5;
  const int nwaves = (gridDim.x * blockDim.x) >> 5;

  for (int f = wave; f < num_fact; f += nwaves) {
    const int r = rels[f];   // wave-uniform loads -> scalarized
    const int t = tails[f];
    const int h = heads[f];

    const float4 v = ((const float4*)(T + (size_t)r * DIM))[lane];

    float* ot = out + (size_t)t * DIM + lane * 4;
    float* oh = out + (size_t)h * DIM + lane * 4;
    unsafeAtomicAdd(ot + 0, v.x);
    unsafeAtomicAdd(ot + 1, v.y);
    unsafeAtomicAdd(ot + 2, v.z);
    unsafeAtomicAdd(ot + 3, v.w);
    unsafeAtomicAdd(oh + 0, v.x);
    unsafeAtomicAdd(oh + 1, v.y);
    unsafeAtomicAdd(oh + 2, v.z);
    unsafeAtomicAdd(oh + 3, v.w);
  }
}

// ---------------------------------------------------------------------------
// Kernel 4: finalize in place: relu + clamp.
// fmaxf(NaN, 0) = 0 reproduces nan->0; fminf(x, CLIP) reproduces +inf->CLIP.
// ---------------------------------------------------------------------------
__global__ void finalize(float4* __restrict__ out, int n4) {
  int i = blockIdx.x * blockDim.x + threadIdx.x;
  if (i < n4) {
    float4 v = out[i];
    v.x = fminf(fmaxf(v.x, 0.f), CLIPF);
    v.y = fminf(fmaxf(v.y, 0.f), CLIPF);
    v.z = fminf(fmaxf(v.z, 0.f), CLIPF);
    v.w = fminf(fmaxf(v.w, 0.f), CLIPF);
    out[i] = v;
  }
}

// ---------------------------------------------------------------------------
// inputs: 0 local_entity(int B*M) 1 batch_heads(int F) 2 batch_rels(int F)
//         3 batch_tails(int F)    4 rel_features(f32 N_REL*D)
//         5 W(f32 D*D)            6 b(f32 D)
// output: f32 B*M*D
// ---------------------------------------------------------------------------
extern "C" void kernel_launch(void* const* d_in, const int* in_sizes, int n_in,
                              void* d_out, int out_size, void* d_ws, size_t ws_size,
                              hipStream_t stream) {
  const int*   heads = (const int*)d_in[1];
  const int*   rels  = (const int*)d_in[2];
  const int*   tails = (const int*)d_in[3];
  const float* relf  = (const float*)d_in[4];
  const float* W     = (const float*)d_in[5];
  const float* bias  = (const float*)d_in[6];
  float*       out   = (float*)d_out;
  float*       T     = (float*)d_ws;          // N_REL * D fp32 = 1 MB scratch
  const int num_fact = in_sizes[1];
  const int n4       = out_size / 4;

  zero_out<<<(n4 + 255) / 256, 256, 0, stream>>>((float4*)out, n4);

  // 125 * 8 = 1000 tiles, 4 waves (128 threads) per block -> 250 blocks
  rel_gemm<<<(N_RELV / 16) * (DIM / 16) / 4, 128, 0, stream>>>(relf, W, bias, T);

  scatter_facts<<<2048, 256, 0, stream>>>(T, rels, heads, tails, out, num_fact);

  finalize<<<(n4 + 255) / 256, 256, 0, stream>>>((float4*)out, n4);
}